// observation_processing_network_85255100825935
// MI455X (gfx1250) — compile-verified
//
#include <hip/hip_runtime.h>
#include <math.h>

#define NODES 2048
#define EDGES 32768
#define ETOT  (EDGES + NODES)

typedef __attribute__((ext_vector_type(2))) float v2f;
typedef __attribute__((ext_vector_type(8))) float v8f;

// ---- ordered-int encoding so float max can use unsigned atomicMax ----
__device__ __forceinline__ unsigned ord_enc(float f) {
  int i = __float_as_int(f);
  return (i >= 0) ? ((unsigned)i | 0x80000000u) : ~(unsigned)i;
}
__device__ __forceinline__ float ord_dec(unsigned u) {
  int i = (u & 0x80000000u) ? (int)(u & 0x7FFFFFFFu) : (int)~u;
  return __int_as_float(i);
}
__device__ __forceinline__ float lrelu(float x) { return x > 0.f ? x : 0.2f * x; }

// ---------------------------------------------------------------- init
__global__ void k_init(const float* __restrict__ x, float* __restrict__ h,
                       float* __restrict__ bval) {
  int n = blockIdx.x * blockDim.x + threadIdx.x;
  if (n == 0) bval[0] = 0.f;
  if (n < NODES) {
    h[n * 3 + 0] = x[n * 3 + 0];
    h[n * 3 + 1] = x[n * 3 + 1];
    h[n * 3 + 2] = x[n * 3 + 2];
  }
}

// ------------------------------------------------------- GAT per-layer
// hW = h @ W (3x3 slice; padded input channels multiply zeros and drop out),
// a_s/a_d, and reset the segment-softmax accumulators.
__global__ void k_gat_pre(const float* __restrict__ h, const float* __restrict__ W,
                          const float* __restrict__ asrc, const float* __restrict__ adst,
                          float* __restrict__ hW, float* __restrict__ a_s,
                          float* __restrict__ a_d, unsigned* __restrict__ bmax,
                          float* __restrict__ bsum, float* __restrict__ bacc) {
  int n = blockIdx.x * blockDim.x + threadIdx.x;
  if (n >= NODES) return;
  float h0 = h[n * 3], h1 = h[n * 3 + 1], h2 = h[n * 3 + 2];
  float w0 = h0 * W[0] + h1 * W[3] + h2 * W[6];
  float w1 = h0 * W[1] + h1 * W[4] + h2 * W[7];
  float w2 = h0 * W[2] + h1 * W[5] + h2 * W[8];
  hW[n * 3] = w0; hW[n * 3 + 1] = w1; hW[n * 3 + 2] = w2;
  a_s[n] = w0 * asrc[0] + w1 * asrc[1] + w2 * asrc[2];
  a_d[n] = w0 * adst[0] + w1 * adst[1] + w2 * adst[2];
  bmax[n] = 0u;  // ordered-int "-infinity"
  bsum[n] = 0.f;
  bacc[n * 3] = 0.f; bacc[n * 3 + 1] = 0.f; bacc[n * 3 + 2] = 0.f;
}

__global__ void k_gat_emax(const int* __restrict__ ei, const float* __restrict__ a_s,
                           const float* __restrict__ a_d, unsigned* __restrict__ bmax) {
  int e = blockIdx.x * blockDim.x + threadIdx.x;
  if (e >= ETOT) return;
  int s = (e < EDGES) ? ei[e] : (e - EDGES);           // self loops appended
  int d = (e < EDGES) ? ei[EDGES + e] : (e - EDGES);
  float t = lrelu(a_s[s] + a_d[d]);
  atomicMax(bmax + d, ord_enc(t));
}

__global__ void k_gat_eacc(const int* __restrict__ ei, const float* __restrict__ a_s,
                           const float* __restrict__ a_d, const unsigned* __restrict__ bmax,
                           const float* __restrict__ hW, float* __restrict__ bsum,
                           float* __restrict__ bacc) {
  int e = blockIdx.x * blockDim.x + threadIdx.x;
  if (e >= ETOT) return;
  int s = (e < EDGES) ? ei[e] : (e - EDGES);
  int d = (e < EDGES) ? ei[EDGES + e] : (e - EDGES);
  float t = lrelu(a_s[s] + a_d[d]);
  float w = expf(t - ord_dec(bmax[d]));
  atomicAdd(bsum + d, w);
  atomicAdd(bacc + d * 3 + 0, w * hW[s * 3 + 0]);
  atomicAdd(bacc + d * 3 + 1, w * hW[s * 3 + 1]);
  atomicAdd(bacc + d * 3 + 2, w * hW[s * 3 + 2]);
}

__global__ void k_gat_post(const float* __restrict__ bacc, const float* __restrict__ bsum,
                           const float* __restrict__ bias, float* __restrict__ h,
                           int do_relu) {
  int n = blockIdx.x * blockDim.x + threadIdx.x;
  if (n >= NODES) return;
  float inv = 1.f / (bsum[n] + 1e-16f);
#pragma unroll
  for (int j = 0; j < 3; ++j) {
    float t = bacc[n * 3 + j] * inv + bias[j];
    if (do_relu) t = t > 0.f ? t : 0.f;
    h[n * 3 + j] = t;
  }
}

// ------------------------------------------------------------- MHA
__global__ void k_qkv(const float* __restrict__ h,
                      const float* __restrict__ Wq, const float* __restrict__ bq,
                      const float* __restrict__ Wk, const float* __restrict__ bk,
                      const float* __restrict__ Wv, const float* __restrict__ bv,
                      float* __restrict__ q, float* __restrict__ k,
                      float* __restrict__ v) {
  int n = blockIdx.x * blockDim.x + threadIdx.x;
  if (n >= NODES) return;
  float h0 = h[n * 3], h1 = h[n * 3 + 1], h2 = h[n * 3 + 2];
#pragma unroll
  for (int j = 0; j < 3; ++j) {
    q[n * 3 + j] = h0 * Wq[j] + h1 * Wq[3 + j] + h2 * Wq[6 + j] + bq[j];
    k[n * 3 + j] = h0 * Wk[j] + h1 * Wk[3 + j] + h2 * Wk[6 + j] + bk[j];
    v[n * 3 + j] = h0 * Wv[j] + h1 * Wv[3 + j] + h2 * Wv[6 + j] + bv[j];
  }
}

// per-head min/max of k -> exact row-max of rank-1 score matrix
__global__ void k_minmax(const float* __restrict__ k, float* __restrict__ kmm) {
  __shared__ float smx[3][128], smn[3][128];
  int t = threadIdx.x;
  float mx[3] = {-3.4e38f, -3.4e38f, -3.4e38f};
  float mn[3] = {3.4e38f, 3.4e38f, 3.4e38f};
  for (int n = t; n < NODES; n += 128)
#pragma unroll
    for (int hh = 0; hh < 3; ++hh) {
      float kv = k[n * 3 + hh];
      mx[hh] = fmaxf(mx[hh], kv);
      mn[hh] = fminf(mn[hh], kv);
    }
#pragma unroll
  for (int hh = 0; hh < 3; ++hh) { smx[hh][t] = mx[hh]; smn[hh][t] = mn[hh]; }
  __syncthreads();
  for (int off = 64; off; off >>= 1) {
    if (t < off)
#pragma unroll
      for (int hh = 0; hh < 3; ++hh) {
        smx[hh][t] = fmaxf(smx[hh][t], smx[hh][t + off]);
        smn[hh][t] = fminf(smn[hh][t], smn[hh][t + off]);
      }
    __syncthreads();
  }
  if (t == 0)
#pragma unroll
    for (int hh = 0; hh < 3; ++hh) { kmm[hh] = smx[hh][0]; kmm[3 + hh] = smn[hh][0]; }
}

// one wave32 per (node, head); streaming softmax, never materializes NxN
__global__ void k_attn(const float* __restrict__ q, const float* __restrict__ k,
                       const float* __restrict__ v, const float* __restrict__ kmm,
                       float* __restrict__ ctx) {
  int gt = blockIdx.x * blockDim.x + threadIdx.x;
  int wg = gt >> 5, lane = gt & 31;
  if (wg >= NODES * 3) return;
  int n = wg / 3, hh = wg % 3;
  float qv = q[n * 3 + hh];
  float M = (qv >= 0.f) ? qv * kmm[hh] : qv * kmm[3 + hh];  // exact row max
  float se = 0.f, sv = 0.f;
  for (int m = lane; m < NODES; m += 32) {
    float p = expf(qv * k[m * 3 + hh] - M);
    se += p;
    sv += p * v[m * 3 + hh];
  }
  for (int off = 16; off; off >>= 1) {
    se += __shfl_xor(se, off, 32);
    sv += __shfl_xor(sv, off, 32);
  }
  if (lane == 0) ctx[n * 3 + hh] = sv / se;
}

// res = ctx @ Wo + bo, plus TransformerConv q/k/v and accumulator reset
__global__ void k_mha_tcpre(const float* __restrict__ ctx,
                            const float* __restrict__ Wo, const float* __restrict__ bo,
                            const float* __restrict__ tWq, const float* __restrict__ tbq,
                            const float* __restrict__ tWk, const float* __restrict__ tbk,
                            const float* __restrict__ tWv, const float* __restrict__ tbv,
                            float* __restrict__ res, float* __restrict__ tq,
                            float* __restrict__ tk, float* __restrict__ tv,
                            unsigned* __restrict__ bmax, float* __restrict__ bsum,
                            float* __restrict__ bacc) {
  int n = blockIdx.x * blockDim.x + threadIdx.x;
  if (n >= NODES) return;
  float c0 = ctx[n * 3], c1 = ctx[n * 3 + 1], c2 = ctx[n * 3 + 2];
  float r[3];
#pragma unroll
  for (int j = 0; j < 3; ++j) {
    r[j] = c0 * Wo[j] + c1 * Wo[3 + j] + c2 * Wo[6 + j] + bo[j];
    res[n * 3 + j] = r[j];
  }
#pragma unroll
  for (int j = 0; j < 3; ++j) {
    tq[n * 3 + j] = r[0] * tWq[j] + r[1] * tWq[3 + j] + r[2] * tWq[6 + j] + tbq[j];
    tk[n * 3 + j] = r[0] * tWk[j] + r[1] * tWk[3 + j] + r[2] * tWk[6 + j] + tbk[j];
    tv[n * 3 + j] = r[0] * tWv[j] + r[1] * tWv[3 + j] + r[2] * tWv[6 + j] + tbv[j];
  }
  bmax[n] = 0u; bsum[n] = 0.f;
  bacc[n * 3] = 0.f; bacc[n * 3 + 1] = 0.f; bacc[n * 3 + 2] = 0.f;
}

#define INV_SQRT3 0.57735026918962576f
__global__ void k_tc_emax(const int* __restrict__ ei, const float* __restrict__ tq,
                          const float* __restrict__ tk, unsigned* __restrict__ bmax) {
  int e = blockIdx.x * blockDim.x + threadIdx.x;
  if (e >= ETOT) return;
  int s = (e < EDGES) ? ei[e] : (e - EDGES);
  int d = (e < EDGES) ? ei[EDGES + e] : (e - EDGES);
  float l = (tq[d * 3] * tk[s * 3] + tq[d * 3 + 1] * tk[s * 3 + 1] +
             tq[d * 3 + 2] * tk[s * 3 + 2]) * INV_SQRT3;
  atomicMax(bmax + d, ord_enc(l));
}

__global__ void k_tc_eacc(const int* __restrict__ ei, const float* __restrict__ tq,
                          const float* __restrict__ tk, const float* __restrict__ tv,
                          const unsigned* __restrict__ bmax, float* __restrict__ bsum,
                          float* __restrict__ bacc) {
  int e = blockIdx.x * blockDim.x + threadIdx.x;
  if (e >= ETOT) return;
  int s = (e < EDGES) ? ei[e] : (e - EDGES);
  int d = (e < EDGES) ? ei[EDGES + e] : (e - EDGES);
  float l = (tq[d * 3] * tk[s * 3] + tq[d * 3 + 1] * tk[s * 3 + 1] +
             tq[d * 3 + 2] * tk[s * 3 + 2]) * INV_SQRT3;
  float w = expf(l - ord_dec(bmax[d]));
  atomicAdd(bsum + d, w);
  atomicAdd(bacc + d * 3 + 0, w * tv[s * 3 + 0]);
  atomicAdd(bacc + d * 3 + 1, w * tv[s * 3 + 1]);
  atomicAdd(bacc + d * 3 + 2, w * tv[s * 3 + 2]);
}

__global__ void k_tc_post(const float* __restrict__ bacc, const float* __restrict__ bsum,
                          const float* __restrict__ res, const float* __restrict__ Ws,
                          const float* __restrict__ bskip, float* __restrict__ outb) {
  int n = blockIdx.x * blockDim.x + threadIdx.x;
  if (n >= NODES) return;
  float inv = 1.f / (bsum[n] + 1e-16f);
  float r0 = res[n * 3], r1 = res[n * 3 + 1], r2 = res[n * 3 + 2];
#pragma unroll
  for (int j = 0; j < 3; ++j)
    outb[n * 3 + j] = bacc[n * 3 + j] * inv + r0 * Ws[j] + r1 * Ws[3 + j] +
                      r2 * Ws[6 + j] + bskip[j];
}

// --------------------------------------------------- MLP head via WMMA
// One wave32 per 16-row tile. Layout per CDNA5 ISA 7.12.2:
//   A 16x4 f32:  lanes 0-15 hold (K0,K1) for row M=lane; lanes 16-31 hold (K2,K3)
//   B 4x16 f32:  lanes 0-15 hold (row K0, row K1) at col N=lane; 16-31 -> (K2,K3)
//   C/D 16x16:   VGPR r = row (r + 8*(lane>=16)), col = lane%16
__global__ __launch_bounds__(32) void
k_mlp(const float* __restrict__ outb, const float* __restrict__ mask,
      const float* __restrict__ W1, const float* __restrict__ b1,
      const float* __restrict__ W2, const float* __restrict__ b2,
      const float* __restrict__ W3, const float* __restrict__ b3,
      float* __restrict__ d_out, float* __restrict__ bval) {
  int tile = blockIdx.x;            // 128 tiles of 16 rows
  int lane = threadIdx.x;
  int col = lane & 15;
  int hi = lane >> 4;               // 0: lanes 0-15, 1: lanes 16-31
  int row0 = tile * 16;

  // ---- layer 1: h1 = relu([out|0,0] @ W1 + b1)  (padded cols 3,4 are zero,
  // so only K=0..3 contribute -> a single 16x16x4 WMMA per tile)
  const float* orow = outb + (row0 + col) * 3;
  v2f a;
  a.x = hi ? orow[2] : orow[0];
  a.y = hi ? 0.f : orow[1];
  int kb = hi ? 2 : 0;
  v2f b;
  b.x = W1[kb * 16 + col];
  b.y = W1[(kb + 1) * 16 + col];
  v8f c1 = {};
  c1 = __builtin_amdgcn_wmma_f32_16x16x4_f32(false, a, false, b, (short)0, c1,
                                             false, false);
  float b1v = b1[col];
  __shared__ float lds[16][17];
#pragma unroll
  for (int r = 0; r < 8; ++r) {
    float t = c1[r] + b1v;
    lds[r + hi * 8][col] = t > 0.f ? t : 0.f;   // stage h1 for re-layout
  }
  __syncthreads();

  // ---- layer 2: h2 = relu(h1 @ W2 + b2) : 2 N-tiles x 4 K-chunks = 8 WMMA
  v2f a2[4];
#pragma unroll
  for (int c = 0; c < 4; ++c) {
    int k0 = 4 * c + 2 * hi;
    a2[c].x = lds[col][k0];
    a2[c].y = lds[col][k0 + 1];
  }
  v8f h2t[2];
#pragma unroll
  for (int t = 0; t < 2; ++t) {
    v8f acc = {};
#pragma unroll
    for (int c = 0; c < 4; ++c) {
      int k0 = 4 * c + 2 * hi;
      v2f bb;
      bb.x = W2[k0 * 32 + t * 16 + col];
      bb.y = W2[(k0 + 1) * 32 + t * 16 + col];
      acc = __builtin_amdgcn_wmma_f32_16x16x4_f32(false, a2[c], false, bb,
                                                  (short)0, acc, false, false);
    }
    float b2v = b2[t * 16 + col];
#pragma unroll
    for (int r = 0; r < 8; ++r) {
      float x = acc[r] + b2v;
      acc[r] = x > 0.f ? x : 0.f;
    }
    h2t[t] = acc;
  }

  // ---- layer 3 (N=1): per-lane partials then 16-lane xor reduction
  float w3a = W3[col], w3b = W3[16 + col];
  float p[8];
#pragma unroll
  for (int r = 0; r < 8; ++r) p[r] = h2t[0][r] * w3a + h2t[1][r] * w3b;
#pragma unroll
  for (int off = 8; off >= 1; off >>= 1)
#pragma unroll
    for (int r = 0; r < 8; ++r) p[r] += __shfl_xor(p[r], off, 32);

  if (col == 0) {                    // lane 0 -> rows 0-7, lane 16 -> rows 8-15
    int base = row0 + hi * 8;
    float b3v = b3[0];
    float rsum = 0.f;
#pragma unroll
    for (int r = 0; r < 8; ++r) {
      float res = p[r] + b3v;
      d_out[base + r] = res * mask[base + r];
      rsum += res;
    }
    atomicAdd(bval, rsum);
  }
}

__global__ void k_final(const float* __restrict__ bval, const float* __restrict__ cW,
                        const float* __restrict__ cb, float* __restrict__ outv) {
  outv[0] = cW[0] * (bval[0] * (1.f / NODES)) + cb[0];
}

// ================================================================ host
extern "C" void kernel_launch(void* const* d_in, const int* in_sizes, int n_in,
                              void* d_out, int out_size, void* d_ws, size_t ws_size,
                              hipStream_t stream) {
  const float* x        = (const float*)d_in[0];
  const int*   ei       = (const int*)d_in[1];
  const float* mask     = (const float*)d_in[2];
  const float* gat_W0   = (const float*)d_in[3];
  const float* gat_W    = (const float*)d_in[4];
  const float* att_src  = (const float*)d_in[5];
  const float* att_dst  = (const float*)d_in[6];
  const float* gat_bias = (const float*)d_in[7];
  const float* mWq = (const float*)d_in[8];  const float* mbq = (const float*)d_in[9];
  const float* mWk = (const float*)d_in[10]; const float* mbk = (const float*)d_in[11];
  const float* mWv = (const float*)d_in[12]; const float* mbv = (const float*)d_in[13];
  const float* mWo = (const float*)d_in[14]; const float* mbo = (const float*)d_in[15];
  const float* tWq = (const float*)d_in[16]; const float* tbq = (const float*)d_in[17];
  const float* tWk = (const float*)d_in[18]; const float* tbk = (const float*)d_in[19];
  const float* tWv = (const float*)d_in[20]; const float* tbv = (const float*)d_in[21];
  const float* tWs = (const float*)d_in[22]; const float* tbs = (const float*)d_in[23];
  const float* W1 = (const float*)d_in[24];  const float* b1 = (const float*)d_in[25];
  const float* W2 = (const float*)d_in[26];  const float* b2 = (const float*)d_in[27];
  const float* W3 = (const float*)d_in[28];  const float* b3 = (const float*)d_in[29];
  const float* cW = (const float*)d_in[30];  const float* cb = (const float*)d_in[31];
  float* out = (float*)d_out;

  float* ws = (float*)d_ws;
  const int N = NODES;
  float*    h    = ws;               // 3N
  float*    hW   = h + 3 * N;        // 3N
  float*    as_  = hW + 3 * N;       // N
  float*    ad_  = as_ + N;          // N
  unsigned* bmax = (unsigned*)(ad_ + N);        // N
  float*    bsum = (float*)bmax + N;            // N
  float*    bacc = bsum + N;         // 3N
  float*    q    = bacc + 3 * N;     // 3N
  float*    k    = q + 3 * N;        // 3N
  float*    v    = k + 3 * N;        // 3N
  float*    ctx  = v + 3 * N;        // 3N
  float*    res  = ctx + 3 * N;      // 3N
  float*    tq   = res + 3 * N;      // 3N
  float*    tk   = tq + 3 * N;       // 3N
  float*    tv   = tk + 3 * N;       // 3N
  float*    outb = tv + 3 * N;       // 3N
  float*    kmm  = outb + 3 * N;     // 6
  float*    bval = kmm + 6;          // 1

  const int TB = 256;
  dim3 gN((N + TB - 1) / TB), gE((ETOT + TB - 1) / TB), blk(TB);

  k_init<<<gN, blk, 0, stream>>>(x, h, bval);

  for (int l = 0; l < 10; ++l) {
    const float* W = (l == 0) ? gat_W0 : (gat_W + (l - 1) * 9);
    k_gat_pre<<<gN, blk, 0, stream>>>(h, W, att_src + l * 3, att_dst + l * 3,
                                      hW, as_, ad_, bmax, bsum, bacc);
    k_gat_emax<<<gE, blk, 0, stream>>>(ei, as_, ad_, bmax);
    k_gat_eacc<<<gE, blk, 0, stream>>>(ei, as_, ad_, bmax, hW, bsum, bacc);
    k_gat_post<<<gN, blk, 0, stream>>>(bacc, bsum, gat_bias + l * 3, h, l < 9);
  }

  k_qkv<<<gN, blk, 0, stream>>>(h, mWq, mbq, mWk, mbk, mWv, mbv, q, k, v);
  k_minmax<<<1, 128, 0, stream>>>(k, kmm);
  k_attn<<<(N * 3 * 32 + TB - 1) / TB, blk, 0, stream>>>(q, k, v, kmm, ctx);
  k_mha_tcpre<<<gN, blk, 0, stream>>>(ctx, mWo, mbo, tWq, tbq, tWk, tbk, tWv, tbv,
                                      res, tq, tk, tv, bmax, bsum, bacc);
  k_tc_emax<<<gE, blk, 0, stream>>>(ei, tq, tk, bmax);
  k_tc_eacc<<<gE, blk, 0, stream>>>(ei, tq, tk, tv, bmax, bsum, bacc);
  k_tc_post<<<gN, blk, 0, stream>>>(bacc, bsum, res, tWs, tbs, outb);

  k_mlp<<<N / 16, 32, 0, stream>>>(outb, mask, W1, b1, W2, b2, W3, b3, out, bval);
  k_final<<<1, 1, 0, stream>>>(bval, cW, cb, out + NODES);
}